// UniNet_MT_V2_72576357367965
// MI455X (gfx1250) — compile-verified
//
#include <hip/hip_runtime.h>
#include <hip/hip_bf16.h>

typedef __attribute__((ext_vector_type(2))) float v2f;
typedef __attribute__((ext_vector_type(8))) float v8f;

#define BB 16
#define NN 16384
#define MM 256

// workspace layout (32-bit words)
#define WSF_COUNTS 0                    // BB*MM floats: cluster counts
#define WSF_SUMS   (BB*MM)              // BB*MM*3 floats: cluster endpoint sums
#define WSF_SL1    (BB*MM*4)            // 1 float: masked smooth-l1 sum
#define WSF_VAR    (BB*MM*4 + 1)        // 1 float: variance accumulator
#define WSF_ZERO_N (BB*MM*4 + 2)        // words to zero-init
#define WSI_IND    (BB*MM*4 + 8)        // BB*NN ints: argmin indices

// ---------------------------------------------------------------------------
// K0: zero the accumulator region of the workspace (harness poisons it).
// ---------------------------------------------------------------------------
__global__ __launch_bounds__(256) void k_zero(float* __restrict__ wsf) {
    int i = blockIdx.x * 256 + threadIdx.x;
    if (i < WSF_ZERO_N) wsf[i] = 0.0f;
}

// ---------------------------------------------------------------------------
// K1: WMMA nearest-contact search + gt_offset + smooth-L1 + segment sums.
// One wave32 per 16-point slab; 8 waves/block, all in the same batch.
// Contacts are pre-transformed once per block into an LDS float4 table
//   q[m] = (-2cx, -2cy, -2cz, ||c||^2  or +INF if m >= total_contacts)
// so each WMMA B operand is a single branchless ds_load_b64 per lane and the
// column-validity mask is baked into the matrix (invalid cols -> +INF).
// grid = BB * (NN/16) / 8 = 2048 blocks of 256 threads.
// ---------------------------------------------------------------------------
__global__ __launch_bounds__(256) void k_argmin_wmma(
    const float* __restrict__ data,      // (B,3,N)
    const float* __restrict__ cp,        // (B,M,3)
    const float* __restrict__ offp,      // (B,3,N)
    const int*   __restrict__ tc_arr,    // (B,)
    const int*   __restrict__ lab_arr,   // (B,)
    float*       __restrict__ out,       // (B,3,N) gt_offset (+ total at end)
    float*       __restrict__ wsf,
    int*         __restrict__ wind)      // (B,N) argmin indices
{
    __shared__ float sq[MM * 4];         // transformed contacts, float4 each

    const int b    = blockIdx.x >> 7;                                // 128 blocks/batch
    const int slab = ((blockIdx.x & 127) << 3) + (threadIdx.x >> 5); // slab in batch
    const int lane = threadIdx.x & 31;

    const int  tc  = tc_arr[b];
    const bool has = tc > 0;
    const float INF = __builtin_inff();

    // ---- stage transformed contacts: one contact per thread ----
    {
        const int m = threadIdx.x;               // 256 threads == MM contacts
        const float cx = cp[b * MM * 3 + m * 3 + 0];
        const float cy = cp[b * MM * 3 + m * 3 + 1];
        const float cz = cp[b * MM * 3 + m * 3 + 2];
        sq[m * 4 + 0] = -2.0f * cx;
        sq[m * 4 + 1] = -2.0f * cy;
        sq[m * 4 + 2] = -2.0f * cz;
        sq[m * 4 + 3] = (m < tc) ? (cx * cx + cy * cy + cz * cz) : INF;
    }
    __syncthreads();

    const int n0   = slab << 4;
    const int bofs = b * 3 * NN;
    const bool lo  = lane < 16;

    // ---- A matrix: 16x4 f32, row = point, cols = [x, y, z, 1] ----
    // lanes 0-15 hold (K=0,K=1) = (x,y); lanes 16-31 hold (K=2,K=3) = (z,1).
    const int n_r = n0 + (lane & 15);
    const float axv = data[bofs + (lo ? 0 : 2 * NN) + n_r]; // x or z (addr select)
    const float ayt = data[bofs + NN + n_r];                // y (all lanes load)
    const float ayv = lo ? ayt : 1.0f;
    v2f A; A.x = axv; A.y = ayv;

    // lane-dependent float offset into sq for the B fragment (2 floats)
    const int qoff = ((lane & 15) << 2) + ((lane >> 4) << 1);

    float bv[8];
    int   bc[8];
#pragma unroll
    for (int i = 0; i < 8; i++) { bv[i] = INF; bc[i] = 0; }

    // ---- 16 column tiles of 16 contacts each ----
#pragma unroll
    for (int ct = 0; ct < MM / 16; ++ct) {
        const float2 qv = *(const float2*)&sq[(ct << 6) + qoff]; // ds_load_b64
        v2f Bm; Bm.x = qv.x; Bm.y = qv.y;
        v8f C = {};
        v8f D = __builtin_amdgcn_wmma_f32_16x16x4_f32(
            false, A, false, Bm, (short)0, C, false, false);
        const int col = (ct << 4) + (lane & 15);
#pragma unroll
        for (int i = 0; i < 8; i++) {
            if (D[i] < bv[i]) { bv[i] = D[i]; bc[i] = col; }  // cols strictly increase
        }
    }

    // ---- reduce (value, col) across the 16 columns in each half-wave ----
#pragma unroll
    for (int i = 0; i < 8; i++) {
#pragma unroll
        for (int mask = 8; mask >= 1; mask >>= 1) {
            float ov = __shfl_xor(bv[i], mask, 16);
            int   oc = __shfl_xor(bc[i], mask, 16);
            if (ov < bv[i] || (ov == bv[i] && oc < bc[i])) { bv[i] = ov; bc[i] = oc; }
        }
    }

    // pick slot j = lane&7 (row = 8*(lane>=16) + j), then redistribute so
    // lanes 0-15 hold rows 0-15 (coalesced epilogue).
    const int j = lane & 7;
    float fv = bv[0];
    int   fc = bc[0];
#pragma unroll
    for (int i = 1; i < 8; i++) { if (j == i) { fv = bv[i]; fc = bc[i]; } }
    (void)fv;
    const int src = (lane < 8) ? lane : lane + 8;   // rows 8-15 live in lanes 16-23
    fc = __shfl(fc, src, 32);

    // pz for row=lane lives in lane^16's A.x
    const float pz_x = __shfl_xor(axv, 16, 32);

    float sl1 = 0.0f;
    if (lane < 16) {
        const int n   = n0 + lane;
        const int ind = has ? fc : 0;
        wind[b * NN + n] = ind;

        const float px = axv;      // x(row=lane) from A fragment
        const float py = ayt;      // y(row=lane)
        const float pz = pz_x;     // z(row=lane) via shuffle

        // recover contact coords from transformed table (exact: *-0.5)
        const float qx = sq[ind * 4 + 0];
        const float qy = sq[ind * 4 + 1];
        const float qz = sq[ind * 4 + 2];
        const float vx = has ? (-0.5f * qx - px) : 0.0f;
        const float vy = has ? (-0.5f * qy - py) : 0.0f;
        const float vz = has ? (-0.5f * qz - pz) : 0.0f;

        out[bofs + n]          = vx;
        out[bofs + NN + n]     = vy;
        out[bofs + 2 * NN + n] = vz;

        const float ox = offp[bofs + n];
        const float oy = offp[bofs + NN + n];
        const float oz = offp[bofs + 2 * NN + n];

        // smooth-L1 vs offset_pred, masked by contact_label == 1
        const float dx = vx - ox, dy = vy - oy, dz = vz - oz;
        const float ax_ = fabsf(dx), ay_ = fabsf(dy), az_ = fabsf(dz);
        float s = (ax_ < 1.0f ? 0.5f * dx * dx : ax_ - 0.5f)
                + (ay_ < 1.0f ? 0.5f * dy * dy : ay_ - 0.5f)
                + (az_ < 1.0f ? 0.5f * dz * dz : az_ - 0.5f);
        if (lab_arr[b] == 1) sl1 = s;

        // segment sums for the variance pass
        if (has) {
            const float ex = px + ox, ey = py + oy, ez = pz + oz;
            float* cnt = wsf + WSF_COUNTS + b * MM + ind;
            float* sm  = wsf + WSF_SUMS + (b * MM + ind) * 3;
            atomicAdd(cnt, 1.0f);
            atomicAdd(sm + 0, ex);
            atomicAdd(sm + 1, ey);
            atomicAdd(sm + 2, ez);
        }
    }

    // wave-level reduction of smooth-L1, one atomic per wave
#pragma unroll
    for (int mask = 16; mask >= 1; mask >>= 1)
        sl1 += __shfl_xor(sl1, mask, 32);
    if (lane == 0 && sl1 != 0.0f)
        atomicAdd(wsf + WSF_SL1, sl1);
}

// ---------------------------------------------------------------------------
// K2: per-point variance contribution ||e - mean[ind]||^2 / count[ind],
// wave-reduced, one atomic per wave. grid = BB*NN/256 = 1024 blocks.
// ---------------------------------------------------------------------------
__global__ __launch_bounds__(256) void k_variance(
    const float* __restrict__ data,
    const float* __restrict__ offp,
    const int*   __restrict__ tc_arr,
    float*       __restrict__ wsf,
    const int*   __restrict__ wind)
{
    const int gid = blockIdx.x * 256 + threadIdx.x;
    const int b   = gid / NN;
    const int n   = gid - b * NN;

    float contrib = 0.0f;
    if (tc_arr[b] > 0) {
        const int   ind  = wind[gid];
        const float cnt  = fmaxf(wsf[WSF_COUNTS + b * MM + ind], 1.0f);
        const float* sm  = wsf + WSF_SUMS + (b * MM + ind) * 3;
        const int bofs   = b * 3 * NN;
        const float ex = data[bofs + n]          + offp[bofs + n];
        const float ey = data[bofs + NN + n]     + offp[bofs + NN + n];
        const float ez = data[bofs + 2 * NN + n] + offp[bofs + 2 * NN + n];
        const float d0 = ex - sm[0] / cnt;
        const float d1 = ey - sm[1] / cnt;
        const float d2 = ez - sm[2] / cnt;
        contrib = (d0 * d0 + d1 * d1 + d2 * d2) / cnt;
    }
#pragma unroll
    for (int mask = 16; mask >= 1; mask >>= 1)
        contrib += __shfl_xor(contrib, mask, 32);
    if ((threadIdx.x & 31) == 0)
        atomicAdd(wsf + WSF_VAR, contrib);
}

// ---------------------------------------------------------------------------
// K3: scalar finalize.
// ---------------------------------------------------------------------------
__global__ void k_final(const int* __restrict__ tc_arr,
                        const int* __restrict__ lab_arr,
                        const float* __restrict__ wsf,
                        float* __restrict__ out)
{
    float nsel1 = 0.0f, nvalid = 0.0f;
    for (int b = 0; b < BB; b++) {
        if (lab_arr[b] == 1) nsel1  += 1.0f;
        if (tc_arr[b]  > 0)  nvalid += 1.0f;
    }
    const float n_sel       = nsel1 * 3.0f * (float)NN;
    const float offset_loss = (n_sel > 0.0f) ? (wsf[WSF_SL1] / fmaxf(n_sel, 1.0f)) : 0.0f;
    const float var_loss    = wsf[WSF_VAR] / (nvalid + 1e-16f);
    out[BB * 3 * NN] = offset_loss * 10.0f + var_loss * 2.0f;
}

// ---------------------------------------------------------------------------
extern "C" void kernel_launch(void* const* d_in, const int* in_sizes, int n_in,
                              void* d_out, int out_size, void* d_ws, size_t ws_size,
                              hipStream_t stream) {
    const float* data = (const float*)d_in[0];   // (B,3,N)
    const float* cp   = (const float*)d_in[1];   // (B,M,3)
    const float* offp = (const float*)d_in[2];   // (B,3,N)
    const int*   tc   = (const int*)d_in[3];     // (B,)
    const int*   lab  = (const int*)d_in[4];     // (B,)
    float* out = (float*)d_out;                  // gt_offset (B*3*N) + total (1)
    float* wsf = (float*)d_ws;
    int*   wind = ((int*)d_ws) + WSI_IND;

    k_zero<<<(WSF_ZERO_N + 255) / 256, 256, 0, stream>>>(wsf);
    k_argmin_wmma<<<BB * (NN / 16) / 8, 256, 0, stream>>>(
        data, cp, offp, tc, lab, out, wsf, wind);
    k_variance<<<BB * NN / 256, 256, 0, stream>>>(data, offp, tc, wsf, wind);
    k_final<<<1, 1, 0, stream>>>(tc, lab, wsf, out);
}